// AttnPool_13065290514620
// MI455X (gfx1250) — compile-verified
//
#include <hip/hip_runtime.h>
#include <hip/hip_bf16.h>

typedef float v2f __attribute__((ext_vector_type(2)));
typedef float v8f __attribute__((ext_vector_type(8)));

#define D_DIM 128
#define C_DIM 256
#define B_SEG 16384
// padded k-pair row stride (in floats) for the LDS image of Wq:
// 288 words == 32 (mod 64 banks) -> half-waves hit disjoint bank halves.
#define WQ_PAIR_STRIDE 288
#define WQ_LDS_FLOATS (64 * WQ_PAIR_STRIDE)   // 18432 floats = 72 KB

// ---------------------------------------------------------------------------
// Kernel 0: ctx_proj[B,128] = ctx_vec[B,256] @ Wk[256,128]   (f32 WMMA)
// One wave per 16x16 output tile. Block = 256 thr = 8 waves = 8 N-tiles.
// ---------------------------------------------------------------------------
__global__ __launch_bounds__(256)
void ctx_proj_kernel(const float* __restrict__ ctx_vec,
                     const float* __restrict__ Wk,
                     float* __restrict__ ctx_proj)
{
    const int lane = threadIdx.x & 31;
    const int wave = threadIdx.x >> 5;     // N-tile 0..7
    const int ln   = lane & 15;
    const int half = lane >> 4;

    const int mBase = blockIdx.x * 16;
    const int nBase = wave * 16;

    const float* rowA = ctx_vec + (size_t)(mBase + ln) * C_DIM;

    v8f acc = {};
#pragma unroll 4
    for (int kt = 0; kt < C_DIM / 4; ++kt) {
        const int k0 = kt * 4 + half * 2;
        v2f a = *(const v2f*)(rowA + k0);
        v2f b;
        b.x = Wk[(size_t)k0       * D_DIM + nBase + ln];
        b.y = Wk[(size_t)(k0 + 1) * D_DIM + nBase + ln];
        acc = __builtin_amdgcn_wmma_f32_16x16x4_f32(
            false, a, false, b, (short)0, acc, false, false);
    }
#pragma unroll
    for (int r = 0; r < 8; ++r) {
        const int m = mBase + r + half * 8;
        ctx_proj[(size_t)m * D_DIM + nBase + ln] = acc[r];
    }
}

// ---------------------------------------------------------------------------
// Kernel 1: scores[N] = tanh(node_x @ Wq + ctx_proj[seg]) @ v
// Block = 256 thr = 8 waves. Wq staged in dynamic LDS, pair-interleaved:
//   element (k,n) at sWq[(k>>1)*288 + 2n + (k&1)]
// so each B fragment is ONE conflict-free ds_load_b64. Each wave owns two
// 16-node M tiles (32 nodes) so every B fragment feeds 2 WMMAs.
// ---------------------------------------------------------------------------
__global__ __launch_bounds__(256)
void score_kernel(const float* __restrict__ node_x,
                  const int*   __restrict__ batch_idx,
                  const float* __restrict__ Wq,
                  const float* __restrict__ vvec,
                  const float* __restrict__ ctx_proj,
                  float* __restrict__ scores)
{
    extern __shared__ float sWq[];        // WQ_LDS_FLOATS (72 KB)

    // stage Wq -> pair-interleaved LDS image (coalesced b32 global reads)
    for (int idx = threadIdx.x; idx < D_DIM * D_DIM; idx += 256) {
        const int k = idx >> 7;           // row (K index)
        const int n = idx & 127;          // col (N index)
        sWq[(k >> 1) * WQ_PAIR_STRIDE + n * 2 + (k & 1)] = Wq[idx];
    }
    __syncthreads();

    const int lane = threadIdx.x & 31;
    const int wave = threadIdx.x >> 5;
    const int ln   = lane & 15;
    const int half = lane >> 4;

    const int nodeBase = blockIdx.x * 256 + wave * 32;   // 2 M tiles per wave

    const float* rowX0 = node_x + (size_t)(nodeBase + ln) * D_DIM;
    const float* rowX1 = rowX0 + (size_t)16 * D_DIM;

    v8f acc[2][8];
#pragma unroll
    for (int t = 0; t < 2; ++t)
#pragma unroll
        for (int nt = 0; nt < 8; ++nt) acc[t][nt] = (v8f){};

    for (int kt = 0; kt < D_DIM / 4; ++kt) {
        const int k0 = kt * 4 + half * 2;             // even
        const v2f a0 = *(const v2f*)(rowX0 + k0);
        const v2f a1 = *(const v2f*)(rowX1 + k0);
        const float* bRow = sWq + (k0 >> 1) * WQ_PAIR_STRIDE;
#pragma unroll
        for (int nt = 0; nt < 8; ++nt) {
            const v2f b = *(const v2f*)(bRow + (nt * 16 + ln) * 2);
            acc[0][nt] = __builtin_amdgcn_wmma_f32_16x16x4_f32(
                false, a0, false, b, (short)0, acc[0][nt], false, false);
            acc[1][nt] = __builtin_amdgcn_wmma_f32_16x16x4_f32(
                false, a1, false, b, (short)0, acc[1][nt], false, false);
        }
    }

    float vreg[8];
#pragma unroll
    for (int nt = 0; nt < 8; ++nt) vreg[nt] = vvec[nt * 16 + ln];

#pragma unroll
    for (int t = 0; t < 2; ++t) {
#pragma unroll
        for (int r = 0; r < 8; ++r) {
            const int mr  = nodeBase + t * 16 + r + half * 8;
            const int seg = batch_idx[mr];
            const float* cp = ctx_proj + (size_t)seg * D_DIM;
            float partial = 0.0f;
#pragma unroll
            for (int nt = 0; nt < 8; ++nt) {
                const float h = tanhf(acc[t][nt][r] + cp[nt * 16 + ln]);
                partial += h * vreg[nt];
            }
#pragma unroll
            for (int off = 1; off < 16; off <<= 1)
                partial += __shfl_xor(partial, off, 16);
            if (ln == 0) scores[mr] = partial;
        }
    }
}

// ---------------------------------------------------------------------------
// Kernel 2: per-segment numerically stable softmax + weighted pooling.
// One wave per segment; batch_idx sorted -> binary search range, then
// max -> sum(exp) -> weighted float4-per-lane streaming accumulation.
// Empty segments write zero rows.
// ---------------------------------------------------------------------------
__global__ __launch_bounds__(32)
void seg_pool_kernel(const float* __restrict__ node_x,
                     const int*   __restrict__ batch_idx,
                     const float* __restrict__ scores,
                     float* __restrict__ out,
                     int N)
{
    const int b    = blockIdx.x;
    const int lane = threadIdx.x;

    int lo = 0, hi = N;
    while (lo < hi) { int mid = (lo + hi) >> 1; if (batch_idx[mid] < b) lo = mid + 1; else hi = mid; }
    const int start = lo;
    hi = N;
    while (lo < hi) { int mid = (lo + hi) >> 1; if (batch_idx[mid] < b + 1) lo = mid + 1; else hi = mid; }
    const int end = lo;

    float m = -__builtin_inff();
    for (int i = start + lane; i < end; i += 32) m = fmaxf(m, scores[i]);
#pragma unroll
    for (int off = 16; off > 0; off >>= 1) m = fmaxf(m, __shfl_xor(m, off, 32));

    float s = 0.0f;
    for (int i = start + lane; i < end; i += 32) s += __expf(scores[i] - m);
#pragma unroll
    for (int off = 16; off > 0; off >>= 1) s += __shfl_xor(s, off, 32);
    const float inv = (end > start) ? 1.0f / s : 0.0f;

    float4 acc = make_float4(0.f, 0.f, 0.f, 0.f);
    int i = start;
    for (; i + 2 <= end; i += 2) {
        const float w0 = __expf(scores[i]     - m) * inv;
        const float w1 = __expf(scores[i + 1] - m) * inv;
        const float4 x0 = ((const float4*)(node_x + (size_t)i * D_DIM))[lane];
        const float4 x1 = ((const float4*)(node_x + (size_t)(i + 1) * D_DIM))[lane];
        acc.x += w0 * x0.x + w1 * x1.x;
        acc.y += w0 * x0.y + w1 * x1.y;
        acc.z += w0 * x0.z + w1 * x1.z;
        acc.w += w0 * x0.w + w1 * x1.w;
    }
    if (i < end) {
        const float w = __expf(scores[i] - m) * inv;
        const float4 x = ((const float4*)(node_x + (size_t)i * D_DIM))[lane];
        acc.x += w * x.x; acc.y += w * x.y; acc.z += w * x.z; acc.w += w * x.w;
    }
    ((float4*)(out + (size_t)b * D_DIM))[lane] = acc;
}

// ---------------------------------------------------------------------------
extern "C" void kernel_launch(void* const* d_in, const int* in_sizes, int n_in,
                              void* d_out, int out_size, void* d_ws, size_t ws_size,
                              hipStream_t stream)
{
    const float* node_x    = (const float*)d_in[0];   // [N,128]
    const int*   batch_idx = (const int*)  d_in[1];   // [N] sorted
    const float* ctx_vec   = (const float*)d_in[2];   // [B,256]
    const float* Wq        = (const float*)d_in[3];   // [128,128]
    const float* Wk        = (const float*)d_in[4];   // [256,128]
    const float* vvec      = (const float*)d_in[5];   // [128]
    float*       out       = (float*)d_out;           // [B,128]

    const int N = in_sizes[0] / D_DIM;                // 1,048,576

    // workspace: ctx_proj [B*128] f32 (8 MB) | scores [N] f32 (4 MB)
    float* ctx_proj = (float*)d_ws;
    float* scores   = ctx_proj + (size_t)B_SEG * D_DIM;

    ctx_proj_kernel<<<B_SEG / 16, 256, 0, stream>>>(ctx_vec, Wk, ctx_proj);

    score_kernel<<<N / 256, 256, WQ_LDS_FLOATS * sizeof(float), stream>>>(
        node_x, batch_idx, Wq, vvec, ctx_proj, scores);

    seg_pool_kernel<<<B_SEG, 32, 0, stream>>>(node_x, batch_idx, scores, out, N);
}